// DynamicGraphEmbedding_76390288327605
// MI455X (gfx1250) — compile-verified
//
#include <hip/hip_runtime.h>
#include <hip/hip_bf16.h>
#include <math.h>

typedef __attribute__((ext_vector_type(2))) float v2f;
typedef __attribute__((ext_vector_type(8))) float v8f;

#define LDT 18          // padded LDS row stride (floats): 72B rows -> b64-aligned, conflict-free
#define BK  16          // K-chunk per LDS stage (4 wmma k-steps)
#define BM  128         // block tile rows
#define BN  64          // block tile cols
#define HBUF ((BM + BN) * LDT)   // one double-buffer half (A tile + B tile)
#define KSEL 308        // k+1 = int(1024*0.3)+1

// ---------------------------------------------------------------------------
// helpers
// ---------------------------------------------------------------------------
__device__ __forceinline__ unsigned f2ord(float f) {
    unsigned u = __float_as_uint(f);
    return (u & 0x80000000u) ? ~u : (u | 0x80000000u);
}

__device__ __forceinline__ v8f wmma4(v2f a, v2f b, v8f c) {
    return __builtin_amdgcn_wmma_f32_16x16x4_f32(false, a, false, b, (short)0, c, false, false);
}

// compute one BK chunk: 2x2 subtiles per wave, 16 wmma
__device__ __forceinline__ void mma_chunk(const float* As, const float* Bs,
                                          int amRow0, int bnRow0, int m, int kb,
                                          v8f& c00, v8f& c01, v8f& c10, v8f& c11) {
#pragma unroll
    for (int ks = 0; ks < BK; ks += 4) {
        v2f a0 = *(const v2f*)&As[(amRow0 + m) * LDT + ks + kb];
        v2f a1 = *(const v2f*)&As[(amRow0 + 16 + m) * LDT + ks + kb];
        v2f b0 = *(const v2f*)&Bs[(bnRow0 + m) * LDT + ks + kb];
        v2f b1 = *(const v2f*)&Bs[(bnRow0 + 16 + m) * LDT + ks + kb];
        c00 = wmma4(a0, b0, c00);
        c01 = wmma4(a0, b1, c01);
        c10 = wmma4(a1, b0, c10);
        c11 = wmma4(a1, b1, c11);
    }
}

__device__ __forceinline__ void store_rowmajor8(float* As, int lr, int lc,
                                                float4 a0, float4 a1) {
    As[lr * LDT + lc + 0] = a0.x; As[lr * LDT + lc + 1] = a0.y;
    As[lr * LDT + lc + 2] = a0.z; As[lr * LDT + lc + 3] = a0.w;
    As[lr * LDT + lc + 4] = a1.x; As[lr * LDT + lc + 5] = a1.y;
    As[lr * LDT + lc + 6] = a1.z; As[lr * LDT + lc + 7] = a1.w;
}

__device__ __forceinline__ void store_rowmajor4(float* Bs, int lr, int lc, float4 b) {
    Bs[lr * LDT + lc + 0] = b.x; Bs[lr * LDT + lc + 1] = b.y;
    Bs[lr * LDT + lc + 2] = b.z; Bs[lr * LDT + lc + 3] = b.w;
}

__device__ __forceinline__ void store_transposed8(float* As, int r, int c8,
                                                  float4 a0, float4 a1) {
    As[(c8 + 0) * LDT + r] = a0.x; As[(c8 + 1) * LDT + r] = a0.y;
    As[(c8 + 2) * LDT + r] = a0.z; As[(c8 + 3) * LDT + r] = a0.w;
    As[(c8 + 4) * LDT + r] = a1.x; As[(c8 + 5) * LDT + r] = a1.y;
    As[(c8 + 6) * LDT + r] = a1.z; As[(c8 + 7) * LDT + r] = a1.w;
}

__device__ __forceinline__ void store_transposed4(float* Bs, int r, int c4, float4 b) {
    Bs[(c4 + 0) * LDT + r] = b.x; Bs[(c4 + 1) * LDT + r] = b.y;
    Bs[(c4 + 2) * LDT + r] = b.z; Bs[(c4 + 3) * LDT + r] = b.w;
}

// ---------------------------------------------------------------------------
// 1) x [B=32, L=256, N=1024] -> xn [(n*32+b)*256 + l]   (node-major rows)
// ---------------------------------------------------------------------------
__global__ __launch_bounds__(256) void k_transpose_x(const float* __restrict__ x,
                                                     float* __restrict__ xn) {
    __shared__ float tile[32][33];
    const int b  = blockIdx.z;
    const int n0 = blockIdx.x * 32;
    const int l0 = blockIdx.y * 32;
    const int tx = threadIdx.x & 31;
    const int ty = threadIdx.x >> 5;   // 0..7
#pragma unroll
    for (int j = 0; j < 4; ++j) {
        const int l = l0 + ty + j * 8;
        tile[ty + j * 8][tx] = x[((size_t)b * 256 + l) * 1024 + n0 + tx];
    }
    __syncthreads();
#pragma unroll
    for (int j = 0; j < 4; ++j) {
        const int n = n0 + ty + j * 8;
        xn[((size_t)n * 32 + b) * 256 + l0 + tx] = tile[tx][ty + j * 8];
    }
}

// ---------------------------------------------------------------------------
// 2) XW = xn @ W   (M=32768, K=256, Ncols=256); 128x64 tiles, double-buffered
// ---------------------------------------------------------------------------
__global__ __launch_bounds__(256) void k_xw(const float* __restrict__ Xn,
                                            const float* __restrict__ W,
                                            float* __restrict__ XW) {
    __shared__ float smem[2 * HBUF];   // 6912 floats

    const int tid  = threadIdx.x;
    const int wave = tid >> 5, lane = tid & 31;
    const int amRow0 = (wave & 3) * 32;
    const int bnRow0 = (wave >> 2) * 32;
    const int m  = lane & 15;
    const int kb = (lane >> 4) << 1;
    const int row0 = blockIdx.y * BM;
    const int col0 = blockIdx.x * BN;
    const int K = 256;

    v8f c00 = {}, c01 = {}, c10 = {}, c11 = {};

    const int lrA = tid >> 1;            // 0..127
    const int lcA = (tid & 1) * 8;       // 0 or 8
    const int brW = tid >> 4;            // 0..15 (k-row)
    const int bcW = (tid & 15) * 4;      // j-chunk

    // stage chunk 0 into buffer 0
    {
        float4 a0 = *(const float4*)&Xn[(size_t)(row0 + lrA) * K + lcA];
        float4 a1 = *(const float4*)&Xn[(size_t)(row0 + lrA) * K + lcA + 4];
        float4 wv = *(const float4*)&W[(size_t)brW * 256 + col0 + bcW];
        store_rowmajor8(smem, lrA, lcA, a0, a1);
        store_transposed4(smem + BM * LDT, brW, bcW, wv);
    }
    __syncthreads();

    int p = 0;
    for (int kt = 0; kt < K; kt += BK) {
        const bool last = (kt + BK >= K);
        float* As = smem + p * HBUF;
        float* Bs = As + BM * LDT;
        float4 a0, a1, wv;
        if (!last) {
            a0 = *(const float4*)&Xn[(size_t)(row0 + lrA) * K + kt + BK + lcA];
            a1 = *(const float4*)&Xn[(size_t)(row0 + lrA) * K + kt + BK + lcA + 4];
            wv = *(const float4*)&W[(size_t)(kt + BK + brW) * 256 + col0 + bcW];
        }
        mma_chunk(As, Bs, amRow0, bnRow0, m, kb, c00, c01, c10, c11);
        if (!last) {
            float* An_ = smem + (p ^ 1) * HBUF;
            store_rowmajor8(An_, lrA, lcA, a0, a1);
            store_transposed4(An_ + BM * LDT, brW, bcW, wv);
            __syncthreads();
        }
        p ^= 1;
    }

    const int nc    = lane & 15;
    const int m_off = (lane >> 4) * 8;
#pragma unroll
    for (int v = 0; v < 8; ++v) {
        const int gr0 = row0 + amRow0 + m_off + v;
        const int gr1 = gr0 + 16;
        const int gc0 = col0 + bnRow0 + nc;
        const int gc1 = gc0 + 16;
        XW[(size_t)gr0 * 256 + gc0] = c00[v];
        XW[(size_t)gr0 * 256 + gc1] = c01[v];
        XW[(size_t)gr1 * 256 + gc0] = c10[v];
        XW[(size_t)gr1 * 256 + gc1] = c11[v];
    }
}

// ---------------------------------------------------------------------------
// 3) in-place L2-normalize each 256-float row of xn (32768 rows)
// ---------------------------------------------------------------------------
__global__ __launch_bounds__(256) void k_normalize_rows(float* __restrict__ xn) {
    const int wave = threadIdx.x >> 5, lane = threadIdx.x & 31;
    const size_t row = (size_t)blockIdx.x * 8 + wave;
    float* p = xn + row * 256;
    float v[8];
    float ss = 0.f;
#pragma unroll
    for (int i = 0; i < 8; ++i) { v[i] = p[lane + 32 * i]; ss += v[i] * v[i]; }
#pragma unroll
    for (int o = 16; o > 0; o >>= 1) ss += __shfl_xor(ss, o, 32);
    const float scale = 1.0f / fmaxf(sqrtf(ss), 1e-12f);
#pragma unroll
    for (int i = 0; i < 8; ++i) p[lane + 32 * i] = v[i] * scale;
}

// ---------------------------------------------------------------------------
// 4) G = (Xnorm @ Xnorm^T) / B   (M=N=1024, K=8192); 128x64 tiles, dbl-buffered
// ---------------------------------------------------------------------------
__global__ __launch_bounds__(256) void k_gram(const float* __restrict__ Xn,
                                              float* __restrict__ G) {
    __shared__ float smem[2 * HBUF];

    const int tid  = threadIdx.x;
    const int wave = tid >> 5, lane = tid & 31;
    const int amRow0 = (wave & 3) * 32;
    const int bnRow0 = (wave >> 2) * 32;
    const int m  = lane & 15;
    const int kb = (lane >> 4) << 1;
    const int row0 = blockIdx.y * BM;
    const int col0 = blockIdx.x * BN;
    const int K = 8192;

    v8f c00 = {}, c01 = {}, c10 = {}, c11 = {};

    const int lrA = tid >> 1;            // 0..127
    const int lcA = (tid & 1) * 8;
    const int lrB = tid >> 2;            // 0..63
    const int lcB = (tid & 3) * 4;

    {
        float4 a0 = *(const float4*)&Xn[(size_t)(row0 + lrA) * K + lcA];
        float4 a1 = *(const float4*)&Xn[(size_t)(row0 + lrA) * K + lcA + 4];
        float4 bv = *(const float4*)&Xn[(size_t)(col0 + lrB) * K + lcB];
        store_rowmajor8(smem, lrA, lcA, a0, a1);
        store_rowmajor4(smem + BM * LDT, lrB, lcB, bv);
    }
    __syncthreads();

    int p = 0;
    for (int kt = 0; kt < K; kt += BK) {
        const bool last = (kt + BK >= K);
        float* As = smem + p * HBUF;
        float* Bs = As + BM * LDT;
        float4 a0, a1, bv;
        if (!last) {
            a0 = *(const float4*)&Xn[(size_t)(row0 + lrA) * K + kt + BK + lcA];
            a1 = *(const float4*)&Xn[(size_t)(row0 + lrA) * K + kt + BK + lcA + 4];
            bv = *(const float4*)&Xn[(size_t)(col0 + lrB) * K + kt + BK + lcB];
        }
        mma_chunk(As, Bs, amRow0, bnRow0, m, kb, c00, c01, c10, c11);
        if (!last) {
            float* An_ = smem + (p ^ 1) * HBUF;
            store_rowmajor8(An_, lrA, lcA, a0, a1);
            store_rowmajor4(An_ + BM * LDT, lrB, lcB, bv);
            __syncthreads();
        }
        p ^= 1;
    }

    const float s   = 1.0f / 32.0f;
    const int nc    = lane & 15;
    const int m_off = (lane >> 4) * 8;
#pragma unroll
    for (int v = 0; v < 8; ++v) {
        const int gr0 = row0 + amRow0 + m_off + v;
        const int gr1 = gr0 + 16;
        const int gc0 = col0 + bnRow0 + nc;
        const int gc1 = gc0 + 16;
        G[(size_t)gr0 * 1024 + gc0] = c00[v] * s;
        G[(size_t)gr0 * 1024 + gc1] = c01[v] * s;
        G[(size_t)gr1 * 1024 + gc0] = c10[v] * s;
        G[(size_t)gr1 * 1024 + gc1] = c11[v] * s;
    }
}

// ---------------------------------------------------------------------------
// 5) per-row top-(k+1) threshold (bit-wise binary search on ordered keys),
//    then A[n,m] = (m!=n && sim>=thr) ? sim : 0   (in place over G)
// ---------------------------------------------------------------------------
__global__ __launch_bounds__(256) void k_topk_mask(float* __restrict__ G) {
    __shared__ int partial[8];
    __shared__ int sh_tot;
    const int n = blockIdx.x;
    const int tid = threadIdx.x, wave = tid >> 5, lane = tid & 31;
    float v[4]; unsigned key[4];
#pragma unroll
    for (int i = 0; i < 4; ++i) {
        v[i]   = G[(size_t)n * 1024 + tid + 256 * i];
        key[i] = f2ord(v[i]);
    }
    unsigned cur = 0u;
    for (int bit = 31; bit >= 0; --bit) {
        const unsigned cand = cur | (1u << bit);
        int c = 0;
#pragma unroll
        for (int i = 0; i < 4; ++i) c += (key[i] >= cand) ? 1 : 0;
#pragma unroll
        for (int o = 16; o > 0; o >>= 1) c += __shfl_xor(c, o, 32);
        if (lane == 0) partial[wave] = c;
        __syncthreads();
        if (tid == 0) {
            int s = 0;
#pragma unroll
            for (int w = 0; w < 8; ++w) s += partial[w];
            sh_tot = s;
        }
        __syncthreads();
        if (sh_tot >= KSEL) cur = cand;
        __syncthreads();
    }
#pragma unroll
    for (int i = 0; i < 4; ++i) {
        const int mm = tid + 256 * i;
        const bool keep = (mm != n) && (key[i] >= cur);
        G[(size_t)n * 1024 + mm] = keep ? v[i] : 0.0f;
    }
}

// ---------------------------------------------------------------------------
// 6) deg[d] = sum_s A[s,d];  dinv[d] = deg>0 ? rsqrt(deg) : 0
// ---------------------------------------------------------------------------
__global__ __launch_bounds__(256) void k_degree(const float* __restrict__ A,
                                                float* __restrict__ dinv) {
    const int d = blockIdx.x * 256 + threadIdx.x;
    float s = 0.f;
    for (int srow = 0; srow < 1024; ++srow) s += A[(size_t)srow * 1024 + d];
    dinv[d] = (s > 0.0f) ? rsqrtf(s) : 0.0f;
}

// ---------------------------------------------------------------------------
// 7) An[s,d] = dinv[s] * A[s,d] * dinv[d]   (in place)
// ---------------------------------------------------------------------------
__global__ __launch_bounds__(256) void k_norm_adj(float* __restrict__ A,
                                                  const float* __restrict__ dinv) {
    const int idx = blockIdx.x * 256 + threadIdx.x;
    const int s = idx >> 10, d = idx & 1023;
    A[idx] = dinv[s] * A[idx] * dinv[d];
}

// ---------------------------------------------------------------------------
// 8) OUT[d,c] = sum_s An[s,d] * XW[s,c]; write transposed + bias:
//    d_out[c*1024 + d] = OUT[d,c] + bias[c & 255]
//    (M=1024 d tiles of 128, Ncols=8192 c tiles of 64, K=1024 s)
// ---------------------------------------------------------------------------
__global__ __launch_bounds__(256) void k_agg(const float* __restrict__ An,
                                             const float* __restrict__ XW,
                                             const float* __restrict__ bias,
                                             float* __restrict__ out) {
    __shared__ float smem[BM * 65];        // 8320 floats >= 2*HBUF = 6912

    const int tid  = threadIdx.x;
    const int wave = tid >> 5, lane = tid & 31;
    const int amRow0 = (wave & 3) * 32;
    const int bnRow0 = (wave >> 2) * 32;
    const int m  = lane & 15;
    const int kb = (lane >> 4) << 1;
    const int row0 = blockIdx.y * BM;      // d
    const int col0 = blockIdx.x * BN;      // c

    v8f c00 = {}, c01 = {}, c10 = {}, c11 = {};

    const int sr  = tid >> 4;              // staged k-row 0..15
    const int sc8 = (tid & 15) * 8;        // A (d) column chunk
    const int sc4 = (tid & 15) * 4;        // B (c) column chunk

    {
        float4 a0 = *(const float4*)&An[(size_t)sr * 1024 + row0 + sc8];
        float4 a1 = *(const float4*)&An[(size_t)sr * 1024 + row0 + sc8 + 4];
        float4 b4 = *(const float4*)&XW[(size_t)sr * 8192 + col0 + sc4];
        store_transposed8(smem, sr, sc8, a0, a1);
        store_transposed4(smem + BM * LDT, sr, sc4, b4);
    }
    __syncthreads();

    int p = 0;
    for (int kt = 0; kt < 1024; kt += BK) {
        const bool last = (kt + BK >= 1024);
        float* As = smem + p * HBUF;
        float* Bs = As + BM * LDT;
        float4 a0, a1, b4;
        if (!last) {
            a0 = *(const float4*)&An[(size_t)(kt + BK + sr) * 1024 + row0 + sc8];
            a1 = *(const float4*)&An[(size_t)(kt + BK + sr) * 1024 + row0 + sc8 + 4];
            b4 = *(const float4*)&XW[(size_t)(kt + BK + sr) * 8192 + col0 + sc4];
        }
        mma_chunk(As, Bs, amRow0, bnRow0, m, kb, c00, c01, c10, c11);
        if (!last) {
            float* An_ = smem + (p ^ 1) * HBUF;
            store_transposed8(An_, sr, sc8, a0, a1);
            store_transposed4(An_ + BM * LDT, sr, sc4, b4);
        }
        __syncthreads();
        p ^= 1;
    }

    // stage 128(d) x 64(c) C tile to LDS, then coalesced transposed write + bias
    float* Cs = smem;                      // Cs[d_local*65 + c_local]
    const int nc    = lane & 15;
    const int m_off = (lane >> 4) * 8;
#pragma unroll
    for (int v = 0; v < 8; ++v) {
        const int dl0 = amRow0 + m_off + v;
        const int dl1 = dl0 + 16;
        Cs[dl0 * 65 + bnRow0 + nc]      = c00[v];
        Cs[dl0 * 65 + bnRow0 + 16 + nc] = c01[v];
        Cs[dl1 * 65 + bnRow0 + nc]      = c10[v];
        Cs[dl1 * 65 + bnRow0 + 16 + nc] = c11[v];
    }
    __syncthreads();
    const int cl  = tid >> 2;              // c_local 0..63
    const int dch = (tid & 3) * 32;        // d chunk (32 each)
    const float bl = bias[(col0 + cl) & 255];
    float* obase = out + (size_t)(col0 + cl) * 1024 + row0 + dch;
#pragma unroll
    for (int j = 0; j < 8; ++j) {
        float4 o;
        o.x = Cs[(dch + 4 * j + 0) * 65 + cl] + bl;
        o.y = Cs[(dch + 4 * j + 1) * 65 + cl] + bl;
        o.z = Cs[(dch + 4 * j + 2) * 65 + cl] + bl;
        o.w = Cs[(dch + 4 * j + 3) * 65 + cl] + bl;
        *(float4*)&obase[4 * j] = o;
    }
}

// ---------------------------------------------------------------------------
// launch
// ---------------------------------------------------------------------------
extern "C" void kernel_launch(void* const* d_in, const int* in_sizes, int n_in,
                              void* d_out, int out_size, void* d_ws, size_t ws_size,
                              hipStream_t stream) {
    const float* x    = (const float*)d_in[0];   // [32, 256, 1024]
    const float* W    = (const float*)d_in[1];   // [256, 256]
    const float* bias = (const float*)d_in[2];   // [256]
    float* out = (float*)d_out;                  // [32, 256, 1024]

    float* ws   = (float*)d_ws;
    float* xn   = ws;                 // 8,388,608 floats (raw xn, then normalized in place)
    float* XW   = ws + 8388608;       // 8,388,608 floats
    float* G    = ws + 16777216;      // 1,048,576 floats (sim -> A -> An in place)
    float* dinv = ws + 17825792;      // 1024 floats
    const size_t need = (size_t)(17825792 + 1024) * sizeof(float);
    if (ws_size < need) return;

    // 1) transpose x -> xn
    k_transpose_x<<<dim3(32, 8, 32), 256, 0, stream>>>(x, xn);
    // 2) XW = xn @ W  (uses raw xn, must precede in-place normalization)
    k_xw<<<dim3(4, 256), 256, 0, stream>>>(xn, W, XW);
    // 3) normalize xn rows in place
    k_normalize_rows<<<4096, 256, 0, stream>>>(xn);
    // 4) G = Xnorm @ Xnorm^T / B
    k_gram<<<dim3(16, 8), 256, 0, stream>>>(xn, G);
    // 5) top-k mask -> A (in place)
    k_topk_mask<<<1024, 256, 0, stream>>>(G);
    // 6) degrees -> dinv
    k_degree<<<4, 256, 0, stream>>>(G, dinv);
    // 7) symmetric normalization -> An (in place)
    k_norm_adj<<<4096, 256, 0, stream>>>(G, dinv);
    // 8) out = An^T @ XW, +bias, transposed store to [B, L, N]
    k_agg<<<dim3(128, 8), 256, 0, stream>>>(G, XW, bias, out);
}